// GaussianMixtureTorch_8100308320888
// MI455X (gfx1250) — compile-verified
//
#include <hip/hip_runtime.h>

typedef __attribute__((ext_vector_type(2))) float v2f;
typedef __attribute__((ext_vector_type(8))) float v8f;

#define B_   4
#define S_   32768
#define F_   32
#define K_   16
#define NBLK 64
#define ROWS_PER_BLK (S_ / NBLK)        /* 512 */
#define NWAVES 8
#define TILES_PER_WAVE (ROWS_PER_BLK / 16 / NWAVES) /* 4 */
#define XPAD 33                          /* conflict-free for stride-32 col reads */
#define RPAD 19                          /* conflict-free for resp^T reads */
#define PART_SZ (2 * K_ + 3 * K_ * F_)   /* 16 bottom + 16 Q0 + 3*512 = 1568 */
#define M1_OFF 32
#define Q1_OFF 544
#define Q2_OFF 1056
#define LOG2PI 1.8378770664093453f
#define REG_COVAR 1e-4f

__device__ inline v8f wmma4(v2f a, v2f b, v8f c) {
  // D = A(16x4 f32) x B(4x16 f32) + C(16x16 f32)
  return __builtin_amdgcn_wmma_f32_16x16x4_f32(false, a, false, b, (short)0, c,
                                               false, false);
}

__device__ inline v8f vzero8() {
  v8f z;
#pragma unroll
  for (int i = 0; i < 8; ++i) z[i] = 0.0f;
  return z;
}

// One iteration's E-step + partial-sum pass.
// grid = (NBLK, B), block = 256 threads (8 wave32).
__global__ __launch_bounds__(256) void em_estep(
    const float* __restrict__ data, const float* __restrict__ means,
    const float* __restrict__ var, float* __restrict__ out_resp,
    float* __restrict__ out_logp, float* __restrict__ partials) {
  __shared__ float sW[64 * 16];   // [c][k]: rows 0..31 = m/v, rows 32..63 = -0.5/v
  __shared__ float sC0[16];
  __shared__ float sX[NWAVES][16 * XPAD];  // per-wave x tile (16 rows x 32 f)
  __shared__ float sR[NWAVES][16 * RPAD];  // per-wave resp tile (16 s x 16 k)
  __shared__ float sAcc[PART_SZ];          // block partial accumulator

  const int blk = blockIdx.x;
  const int b = blockIdx.y;
  const int tid = threadIdx.x;
  const int lane = tid & 31;
  const int wv = tid >> 5;
  const int half = lane >> 4;   // 0 for lanes 0-15, 1 for lanes 16-31
  const int l16 = lane & 15;

  // ---- Build weight matrix W and constants c0 from means/var ----
  for (int idx = tid; idx < K_ * F_; idx += 256) {
    int k = idx & 15, f = idx >> 4;
    float m = means[(b * K_ + k) * F_ + f];
    float v = var[(b * K_ + k) * F_ + f];
    sW[f * 16 + k] = m / v;                 // coefficient on x
    sW[(F_ + f) * 16 + k] = -0.5f / v;      // coefficient on x^2
  }
  if (tid < K_) {
    float acc = 0.f;
    for (int f = 0; f < F_; ++f) {
      float m = means[(b * K_ + tid) * F_ + f];
      float v = var[(b * K_ + tid) * F_ + f];
      acc += m * m / v + __logf(v);
    }
    sC0[tid] = -0.5f * (acc + (float)F_ * LOG2PI);
  }
  __syncthreads();

  const float c0r = sC0[l16];
  const unsigned ldsXbase = (unsigned)(size_t)(&sX[wv][0]);

  // Per-wave WMMA accumulators: M1 (resp^T X), Q1 (resp2^T X), Q2 (resp2^T X^2)
  v8f accM1a = vzero8(), accM1b = vzero8();
  v8f accQ1a = vzero8(), accQ1b = vzero8();
  v8f accQ2a = vzero8(), accQ2b = vzero8();
  float accBot = 0.f, accQ0 = 0.f;

  const int sblk = blk * ROWS_PER_BLK;

#pragma unroll 1
  for (int t = 0; t < TILES_PER_WAVE; ++t) {
    const int tile = wv + t * NWAVES;
    const int sbase = sblk + tile * 16;
    const float* gX = data + ((size_t)b * S_ + sbase) * F_;
    if (t + 1 < TILES_PER_WAVE)
      __builtin_prefetch(data + ((size_t)b * S_ + sbase + NWAVES * 16) * F_, 0, 0);

    // ---- Stage 16x32 x-tile straight into LDS via async copy (no VGPR trip).
    // Each lane moves 4x16 bytes; tracked with ASYNCcnt.
#pragma unroll
    for (int q = 0; q < 4; ++q) {
      const int flat = q * 128 + lane * 4;
      const int r = flat >> 5, c = flat & 31;
      const unsigned ldsoff = ldsXbase + (unsigned)((r * XPAD + c) * 4);
      const float* gp = gX + r * F_ + c;
      asm volatile("global_load_async_to_lds_b128 %0, %1, off"
                   :
                   : "v"(ldsoff), "v"(gp)
                   : "memory");
    }
    asm volatile("s_wait_asynccnt 0x0" ::: "memory");

    // ---- Preload forward A fragments: lane needs row l16, cols {4s+2h,+1}.
    // Steps 8..15 reuse the same columns squared (x^2 block of [x,x^2]).
    v2f Ax[8];
#pragma unroll
    for (int s = 0; s < 8; ++s) {
      const int f = s * 4 + half * 2;
      Ax[s][0] = sX[wv][l16 * XPAD + f];
      Ax[s][1] = sX[wv][l16 * XPAD + f + 1];
    }

    // ---- Forward GEMM: logp(16x16) = [x,x^2](16x64) * W(64x16) + c0.
    // Two independent accumulator chains for XDL overlap.
    v8f D0, D1;
#pragma unroll
    for (int i = 0; i < 8; ++i) { D0[i] = c0r; D1[i] = 0.f; }
#pragma unroll
    for (int s = 0; s < 8; ++s) {
      const int c64 = s * 4 + half * 2;
      v2f Bv; Bv[0] = sW[c64 * 16 + l16]; Bv[1] = sW[(c64 + 1) * 16 + l16];
      if (s & 1) D1 = wmma4(Ax[s], Bv, D1);
      else       D0 = wmma4(Ax[s], Bv, D0);
    }
#pragma unroll
    for (int s = 0; s < 8; ++s) {
      const int c64 = 32 + s * 4 + half * 2;
      const v2f Ax2 = Ax[s] * Ax[s];
      v2f Bv; Bv[0] = sW[c64 * 16 + l16]; Bv[1] = sW[(c64 + 1) * 16 + l16];
      if (s & 1) D1 = wmma4(Ax2, Bv, D1);
      else       D0 = wmma4(Ax2, Bv, D0);
    }
    v8f D;
#pragma unroll
    for (int i = 0; i < 8; ++i) D[i] = D0[i] + D1[i];

    // ---- responsibility: prob=exp(logp)+1e-8, L1 normalize over k ----
    float respv[8];
#pragma unroll
    for (int j = 0; j < 8; ++j) {
      float p = __expf(D[j]) + 1e-8f;
      float s = p;
      s += __shfl_xor(s, 1);
      s += __shfl_xor(s, 2);
      s += __shfl_xor(s, 4);
      s += __shfl_xor(s, 8);   // sum over 16 k-lanes within each half
      float r = p / s;
      respv[j] = r;
      accBot += r;
      accQ0 += r * r;
    }

    // ---- write resp / log_prob outputs (valid after last iteration) ----
    const size_t obase = ((size_t)b * S_ + sbase) * K_;
#pragma unroll
    for (int j = 0; j < 8; ++j) {
      const int row = j + 8 * half;
      out_logp[obase + row * K_ + l16] = D[j];
      out_resp[obase + row * K_ + l16] = respv[j];
    }

    // ---- stage resp^T tile into LDS ----
#pragma unroll
    for (int j = 0; j < 8; ++j)
      sR[wv][(j + 8 * half) * RPAD + l16] = respv[j];

    // ---- Preload M-step fragments (A = resp^T rows, B = x rows) ----
    v2f Rg[4], Xa[4], Xb[4];
#pragma unroll
    for (int g = 0; g < 4; ++g) {
      const int s0 = g * 4 + half * 2;
      Rg[g][0] = sR[wv][s0 * RPAD + l16];
      Rg[g][1] = sR[wv][(s0 + 1) * RPAD + l16];
      Xa[g][0] = sX[wv][s0 * XPAD + l16];
      Xa[g][1] = sX[wv][(s0 + 1) * XPAD + l16];
      Xb[g][0] = sX[wv][s0 * XPAD + 16 + l16];
      Xb[g][1] = sX[wv][(s0 + 1) * XPAD + 16 + l16];
    }

    // ---- M-step GEMMs: 24 WMMAs across 6 independent accumulators ----
#pragma unroll
    for (int g = 0; g < 4; ++g) {
      const v2f R2 = Rg[g] * Rg[g];
      const v2f X2a = Xa[g] * Xa[g];
      const v2f X2b = Xb[g] * Xb[g];
      accM1a = wmma4(Rg[g], Xa[g], accM1a);
      accM1b = wmma4(Rg[g], Xb[g], accM1b);
      accQ1a = wmma4(R2, Xa[g], accQ1a);
      accQ1b = wmma4(R2, Xb[g], accQ1b);
      accQ2a = wmma4(R2, X2a, accQ2a);
      accQ2b = wmma4(R2, X2b, accQ2b);
    }
  }

  // combine lane halves for bottom/Q0 (each k now totalled in both halves)
  accBot += __shfl_xor(accBot, 16);
  accQ0 += __shfl_xor(accQ0, 16);

  // ---- deterministic cross-wave reduction (fixed wave order) ----
  for (int w = 0; w < NWAVES; ++w) {
    if (wv == w) {
      const bool first = (w == 0);
      if (lane < 16) {
        sAcc[lane] = (first ? 0.f : sAcc[lane]) + accBot;
        sAcc[16 + lane] = (first ? 0.f : sAcc[16 + lane]) + accQ0;
      }
#pragma unroll
      for (int j = 0; j < 8; ++j) {
        const int k = j + 8 * half;
        const int ia = k * F_ + l16;
        const int ib = k * F_ + 16 + l16;
        if (first) {
          sAcc[M1_OFF + ia] = accM1a[j]; sAcc[M1_OFF + ib] = accM1b[j];
          sAcc[Q1_OFF + ia] = accQ1a[j]; sAcc[Q1_OFF + ib] = accQ1b[j];
          sAcc[Q2_OFF + ia] = accQ2a[j]; sAcc[Q2_OFF + ib] = accQ2b[j];
        } else {
          sAcc[M1_OFF + ia] += accM1a[j]; sAcc[M1_OFF + ib] += accM1b[j];
          sAcc[Q1_OFF + ia] += accQ1a[j]; sAcc[Q1_OFF + ib] += accQ1b[j];
          sAcc[Q2_OFF + ia] += accQ2a[j]; sAcc[Q2_OFF + ib] += accQ2b[j];
        }
      }
    }
    __syncthreads();
  }

  // ---- write block partial ----
  float* gp = partials + ((size_t)b * NBLK + blk) * PART_SZ;
  for (int i = tid; i < PART_SZ; i += 256) gp[i] = sAcc[i];
}

// Finalize one iteration: reduce partials in fixed order, update means/var.
// grid = B, block = 512 (one thread per (k,f)).
__global__ __launch_bounds__(512) void em_mstep(
    const float* __restrict__ partials, float* __restrict__ wmeans,
    float* __restrict__ wvar, float* __restrict__ out_means,
    float* __restrict__ out_cov, float* __restrict__ out_prior, int last) {
  __shared__ float sBot[16], sQ0[16];
  const int b = blockIdx.x;
  const int t = threadIdx.x;
  const float* gp = partials + (size_t)b * NBLK * PART_SZ;

  if (t < 32) {
    float s = 0.f;
    for (int p = 0; p < NBLK; ++p) s += gp[(size_t)p * PART_SZ + t];
    if (t < 16) sBot[t] = s; else sQ0[t - 16] = s;
  }
  float s1 = 0.f, s2 = 0.f, s3 = 0.f;
  for (int p = 0; p < NBLK; ++p) {
    const float* q = gp + (size_t)p * PART_SZ;
    s1 += q[M1_OFF + t];
    s2 += q[Q1_OFF + t];
    s3 += q[Q2_OFF + t];
  }
  __syncthreads();

  const int k = t >> 5, f = t & 31;
  const float bot = sBot[k];
  const float m = s1 / bot;
  float v = (s3 - 2.f * m * s2 + m * m * sQ0[k]) / bot;
  v = fmaxf(v, REG_COVAR);

  const int mi = (b * K_ + k) * F_ + f;
  wmeans[mi] = m;
  wvar[mi] = v;

  if (last) {
    out_means[mi] = m;
    float* cr = out_cov + (size_t)mi * F_;
    for (int j = 0; j < F_; ++j) cr[j] = (j == f) ? v : 0.f;
    if (t < K_) out_prior[b * K_ + t] = sBot[t];
  }
}

extern "C" void kernel_launch(void* const* d_in, const int* in_sizes, int n_in,
                              void* d_out, int out_size, void* d_ws,
                              size_t ws_size, hipStream_t stream) {
  const float* data = (const float*)d_in[0];    // (B,S,F)
  const float* means0 = (const float*)d_in[1];  // (B,K,F)
  const float* cov0 = (const float*)d_in[2];    // (B,K,F)

  float* out = (float*)d_out;
  float* out_resp = out;                           // B*S*K = 2097152
  float* out_logp = out + 2097152;                 // B*S*K
  float* out_means = out + 4194304;                // B*K*F = 2048
  float* out_cov = out + 4196352;                  // B*K*F*F = 65536
  float* out_prior = out + 4261888;                // B*K = 64

  float* ws = (float*)d_ws;
  float* partials = ws;                            // B*NBLK*PART_SZ = 401408
  float* wmeans = ws + (size_t)B_ * NBLK * PART_SZ;
  float* wvar = wmeans + B_ * K_ * F_;

  for (int it = 0; it < 5; ++it) {
    const float* mp = (it == 0) ? means0 : wmeans;
    const float* vp = (it == 0) ? cov0 : wvar;
    em_estep<<<dim3(NBLK, B_), 256, 0, stream>>>(data, mp, vp, out_resp,
                                                 out_logp, partials);
    em_mstep<<<B_, 512, 0, stream>>>(partials, wmeans, wvar, out_means,
                                     out_cov, out_prior, (it == 4) ? 1 : 0);
  }
}